// DetectionLayer_19954418057797
// MI455X (gfx1250) — compile-verified
//
#include <hip/hip_runtime.h>
#include <hip/hip_bf16.h>
#include <float.h>

// Problem constants (match reference)
#define BB      16
#define NN      98304
#define CC      20
#define ROWLEN  24          // 4 + C
#define KPRE    200
#define KPAD    208         // 13 * 16
#define NT      13          // 16-wide tiles covering 208
#define MAXPC   50
#define MAXDET  50
#define IOU_T   0.1f
#define CONF_T  0.5f
#define NPAIR   (BB * CC)   // 320
#define NCAND   (CC * MAXPC)// 1000

typedef float v2f __attribute__((ext_vector_type(2)));
typedef float v8f __attribute__((ext_vector_type(8)));

__device__ __forceinline__ unsigned sortable(float f) {
  unsigned u = __float_as_uint(f);
  return u ^ ((u >> 31) ? 0xFFFFFFFFu : 0x80000000u);
}

// ---------------------------------------------------------------------------
// Kernel 1: per (image, class) exact top-200 of N=98304 logits.
// 16-bit radix select: two histogram passes over a 65536-bin LDS histogram
// (dynamic LDS; needs CDNA5's 320KB WGP LDS), one fill pass, bitonic sort.
// predictions stays L2-resident (151MB < 192MB L2) so the 3 scans are cheap.
// ---------------------------------------------------------------------------
extern __shared__ unsigned dynHist[];   // 65536 u32 = 256 KB dynamic LDS

__global__ __launch_bounds__(256) void topk_kernel(
    const float* __restrict__ pred,   // (B, N, 24)
    float* __restrict__ tkScore,      // (NPAIR, KPRE) logits, sorted desc
    int*   __restrict__ tkIdx)        // (NPAIR, KPRE)
{
  const int pair = blockIdx.x;          // b*C + c
  const int b = pair / CC, c = pair % CC;
  const int tid = threadIdx.x;
  const size_t base = (size_t)b * NN * ROWLEN + 4 + c;

  __shared__ unsigned bsum[256];
  __shared__ unsigned s_b16, s_need, s_above, s_T;
  __shared__ unsigned s_cntG, s_cntE;
  __shared__ float csc[256];
  __shared__ int   cidx[256];

  if (tid == 0) { s_need = KPRE; s_above = 0u; }

  // ---- pass 0: histogram of top 16 key bits ----
  for (int q = tid; q < 65536; q += 256) dynHist[q] = 0u;
  __syncthreads();
  for (int n = tid; n < NN; n += 256) {
    const float* ap = pred + base + (size_t)n * ROWLEN;
    __builtin_prefetch(ap + 128 * ROWLEN, 0, 0);      // global_prefetch_b8
    unsigned key = sortable(*ap);
    atomicAdd(&dynHist[key >> 16], 1u);
  }
  __syncthreads();
  {
    unsigned loc = 0;
    for (int q = 0; q < 256; ++q) loc += dynHist[tid * 256 + q];
    bsum[tid] = loc;
  }
  __syncthreads();
  if (tid == 0) {
    unsigned need = s_need, acc = 0; int blk = 0, bin = 0;
    for (int i = 255; i >= 0; --i) {
      if (acc + bsum[i] >= need) { blk = i; break; }
      acc += bsum[i];
    }
    for (int i = blk * 256 + 255; i >= blk * 256; --i) {
      if (acc + dynHist[i] >= need) { bin = i; break; }
      acc += dynHist[i];
    }
    s_b16 = (unsigned)bin;
    s_need = need - acc;
    s_above = acc;
  }
  __syncthreads();

  // ---- pass 1: histogram of low 16 bits, restricted to selected hi bits ----
  const unsigned b16 = s_b16;
  for (int q = tid; q < 65536; q += 256) dynHist[q] = 0u;
  __syncthreads();
  for (int n = tid; n < NN; n += 256) {
    const float* ap = pred + base + (size_t)n * ROWLEN;
    __builtin_prefetch(ap + 128 * ROWLEN, 0, 0);
    unsigned key = sortable(*ap);
    if ((key >> 16) == b16) atomicAdd(&dynHist[key & 0xFFFFu], 1u);
  }
  __syncthreads();
  {
    unsigned loc = 0;
    for (int q = 0; q < 256; ++q) loc += dynHist[tid * 256 + q];
    bsum[tid] = loc;
  }
  __syncthreads();
  if (tid == 0) {
    unsigned need = s_need;
    unsigned above0 = s_above;
    int blk = 0, bin = 0;
    unsigned a2 = 0;
    for (int i = 255; i >= 0; --i) {
      if (a2 + bsum[i] >= need) { blk = i; break; }
      a2 += bsum[i];
    }
    for (int i = blk * 256 + 255; i >= blk * 256; --i) {
      if (a2 + dynHist[i] >= need) { bin = i; break; }
      a2 += dynHist[i];
    }
    s_T = (b16 << 16) | (unsigned)bin;
    s_need = need - a2;
    s_above = above0 + a2;
  }
  __syncthreads();

  const unsigned T = s_T;
  const unsigned above = s_above;
  const unsigned need = s_need;

  // ---- fill pass: gather exactly 200 candidates ----
  csc[tid] = -FLT_MAX; cidx[tid] = 0x7FFFFFFF;
  if (tid == 0) { s_cntG = 0u; s_cntE = 0u; }
  __syncthreads();
  for (int n = tid; n < NN; n += 256) {
    float v = pred[base + (size_t)n * ROWLEN];
    unsigned key = sortable(v);
    if (key > T) {
      unsigned p = atomicAdd(&s_cntG, 1u);
      csc[p] = v; cidx[p] = n;
    } else if (key == T) {
      unsigned e = atomicAdd(&s_cntE, 1u);
      if (e < need) { csc[above + e] = v; cidx[above + e] = n; }
    }
  }
  __syncthreads();

  // ---- bitonic sort 256 entries: score desc, index asc (lax.top_k order) ----
  for (unsigned k = 2; k <= 256; k <<= 1) {
    for (unsigned j = k >> 1; j > 0; j >>= 1) {
      unsigned ixj = tid ^ j;
      if (ixj > (unsigned)tid) {
        float sa = csc[tid], sb = csc[ixj];
        int   ia = cidx[tid], ib = cidx[ixj];
        bool aFirst  = (sa > sb) || (sa == sb && ia < ib);
        bool dirDesc = ((tid & k) == 0);
        if (aFirst != dirDesc) {
          csc[tid] = sb; csc[ixj] = sa;
          cidx[tid] = ib; cidx[ixj] = ia;
        }
      }
      __syncthreads();
    }
  }

  if (tid < KPRE) {
    tkScore[pair * KPRE + tid] = csc[tid];
    tkIdx[pair * KPRE + tid]   = cidx[tid];
  }
}

// ---------------------------------------------------------------------------
// Kernel 2: per (image, class) NMS over the sorted top-200.
//  * scattered 16B pred-box / anchor rows fetched via CDNA5 async DMA to LDS
//  * pairwise (area_i + area_j) matrix via V_WMMA_F32_16X16X4_F32
//  * division-free IoU predicate: inter > t*max(union,eps)
//  * boxes packed as float4 in LDS -> single ds_load_b128 per box
//  * iou>thresh predicate packed with wave32 ballot into 16-bit chunks
//  * greedy suppression as a 13-word bitmask sweep
// ---------------------------------------------------------------------------
__global__ __launch_bounds__(256) void nms_kernel(
    const float* __restrict__ pred,     // (B, N, 24)
    const float* __restrict__ anchors,  // (N, 4)
    const float* __restrict__ tkScore,
    const int*   __restrict__ tkIdx,
    float* __restrict__ nmsScore,       // (NPAIR, 50)
    float* __restrict__ nmsBox)         // (NPAIR, 50, 4)
{
  const int pair = blockIdx.x;
  const int b = pair / CC;
  const int tid  = threadIdx.x;
  const int lane = tid & 31;
  const int wave = tid >> 5;

  __shared__ float4 predStage[256];
  __shared__ float4 anchStage[256];
  __shared__ float4 sbox[KPAD];         // {x1, y1, x2, y2}
  __shared__ float  sarea[KPAD], ssc[KPAD];
  __shared__ unsigned sup[KPAD][NT];

  int   idx   = 0;
  float logit = -FLT_MAX;
  if (tid < KPRE) {
    idx   = tkIdx[pair * KPRE + tid];
    logit = tkScore[pair * KPRE + tid];
  }

  // CDNA5 async global->LDS DMA for the scattered 16B gathers
  {
    unsigned ldsP = (unsigned)(unsigned long long)&predStage[tid];
    unsigned ldsA = (unsigned)(unsigned long long)&anchStage[tid];
    unsigned long long gP = (unsigned long long)(const void*)
        (pred + (size_t)b * NN * ROWLEN + (size_t)idx * ROWLEN);
    unsigned long long gA = (unsigned long long)(const void*)
        (anchors + (size_t)idx * 4);
    asm volatile("global_load_async_to_lds_b128 %0, %1, off"
                 :: "v"(ldsP), "v"(gP) : "memory");
    asm volatile("global_load_async_to_lds_b128 %0, %1, off"
                 :: "v"(ldsA), "v"(gA) : "memory");
    asm volatile("s_wait_asynccnt 0x0" ::: "memory");
  }
  __syncthreads();

  if (tid < KPAD) {
    if (tid < KPRE) {
      float4 p = predStage[tid];
      float4 a = anchStage[tid];
      float cx = p.x * a.z + a.x;
      float cy = p.y * a.w + a.y;
      float w  = expf(p.z) * a.z;
      float h  = expf(p.w) * a.w;
      float4 bx;
      bx.x = cx - 0.5f * w; bx.y = cy - 0.5f * h;
      bx.z = cx + 0.5f * w; bx.w = cy + 0.5f * h;
      sbox[tid] = bx;
      sarea[tid] = (bx.z - bx.x) * (bx.w - bx.y);
      ssc[tid] = 1.0f / (1.0f + expf(-logit));
    } else {
      sbox[tid] = make_float4(0.f, 0.f, 0.f, 0.f);
      sarea[tid] = 0.f; ssc[tid] = -1.0f;
    }
  }
  __syncthreads();

  // pairwise IoU tiles; wave-cooperative; EXEC all-ones at the WMMA
  const v8f czero = {0.f, 0.f, 0.f, 0.f, 0.f, 0.f, 0.f, 0.f};
  for (int t = wave; t < NT * NT; t += 8) {
    const int ti = t / NT, tj = t % NT;
    const int m = lane & 15, half = lane >> 4;

    // A (16x4): K0 = area_row, K1 = 1 (lanes 0-15); K2=K3=0 (lanes 16-31)
    v2f A; A.x = half ? 0.f : sarea[ti * 16 + m];
           A.y = half ? 0.f : 1.0f;
    // B (4x16): K0 row = 1 (lanes 0-15), K1 row = area_col (lanes 16-31)
    v2f Bv; Bv.x = half ? sarea[tj * 16 + m] : 1.0f;
            Bv.y = 0.f;

    // C[i][j] = area_i + area_j  ->  v_wmma_f32_16x16x4_f32
    v8f csum = __builtin_amdgcn_wmma_f32_16x16x4_f32(
        false, A, false, Bv, (short)0, czero, false, false);

    const int j = tj * 16 + m;
    const float4 bj = sbox[j];                 // one ds_load_b128
#pragma unroll
    for (int v = 0; v < 8; ++v) {
      const int i = ti * 16 + v + 8 * half;    // C-tile VGPR layout
      const float4 bi = sbox[i];               // one ds_load_b128
      float iw = fminf(bi.z, bj.z) - fmaxf(bi.x, bj.x);
      float ih = fminf(bi.w, bj.w) - fmaxf(bi.y, bj.y);
      float inter = fmaxf(iw, 0.f) * fmaxf(ih, 0.f);
      float uni = csum[v] - inter;
      // iou > t  <=>  inter > t * max(union, eps)   (divisor > 0)
      bool supPred = inter > IOU_T * fmaxf(uni, 1e-8f);
      unsigned mask = __builtin_amdgcn_ballot_w32(supPred);
      if (lane == 0) {
        sup[ti * 16 + v][tj]     = mask & 0xFFFFu;   // rows from lanes 0-15
        sup[ti * 16 + 8 + v][tj] = mask >> 16;       // rows from lanes 16-31
      }
    }
  }
  __syncthreads();

  // greedy sequential NMS + compaction (tiny; single thread)
  if (tid == 0) {
    unsigned kc[NT];
    for (int ch = 0; ch < NT; ++ch) {
      unsigned w = 0;
      for (int q = 0; q < 16; ++q)
        if (ssc[ch * 16 + q] > CONF_T) w |= 1u << q;
      kc[ch] = w;                       // keep starts as `valid`
    }
    for (int i = 0; i < KPRE; ++i) {
      const int ci = i >> 4, bi = i & 15;
      if ((kc[ci] >> bi) & 1u) {
        for (int ch = ci; ch < NT; ++ch) {
          unsigned m = sup[i][ch];
          if (ch == ci) m &= ~((2u << bi) - 1u);   // suppress only j > i
          kc[ch] &= ~m;
        }
      }
    }
    int cnt = 0;
    for (int i = 0; i < KPRE && cnt < MAXPC; ++i) {
      if ((kc[i >> 4] >> (i & 15)) & 1u) {
        const int o = pair * MAXPC + cnt;
        float4 bx = sbox[i];
        nmsScore[o] = ssc[i];
        nmsBox[o * 4 + 0] = bx.x;
        nmsBox[o * 4 + 1] = bx.y;
        nmsBox[o * 4 + 2] = bx.z;
        nmsBox[o * 4 + 3] = bx.w;
        ++cnt;
      }
    }
    for (; cnt < MAXPC; ++cnt) {
      const int o = pair * MAXPC + cnt;
      nmsScore[o] = -1.0f;
      nmsBox[o * 4 + 0] = 0.f; nmsBox[o * 4 + 1] = 0.f;
      nmsBox[o * 4 + 2] = 0.f; nmsBox[o * 4 + 3] = 0.f;
    }
  }
}

// ---------------------------------------------------------------------------
// Kernel 3: per image, merge 20*50 candidates -> top-50 + class ids + count.
// ---------------------------------------------------------------------------
__global__ __launch_bounds__(1024) void combine_kernel(
    const float* __restrict__ nmsScore,
    const float* __restrict__ nmsBox,
    float* __restrict__ out)
{
  const int b = blockIdx.x;
  const int tid = threadIdx.x;

  __shared__ float ssc[1024];
  __shared__ int   sid[1024];

  if (tid < NCAND) { ssc[tid] = nmsScore[b * NCAND + tid]; sid[tid] = tid; }
  else             { ssc[tid] = -FLT_MAX;                  sid[tid] = tid; }
  __syncthreads();

  for (unsigned k = 2; k <= 1024; k <<= 1) {
    for (unsigned j = k >> 1; j > 0; j >>= 1) {
      unsigned ixj = tid ^ j;
      if (ixj > (unsigned)tid) {
        float sa = ssc[tid], sb = ssc[ixj];
        int   ia = sid[tid], ib = sid[ixj];
        bool aFirst  = (sa > sb) || (sa == sb && ia < ib);
        bool dirDesc = ((tid & k) == 0);
        if (aFirst != dirDesc) {
          ssc[tid] = sb; ssc[ixj] = sa;
          sid[tid] = ib; sid[ixj] = ia;
        }
      }
      __syncthreads();
    }
  }

  if (tid < MAXDET) {
    float sc = ssc[tid];
    int   id = sid[tid];
    bool valid = (sc > 0.f) && (id < NCAND);
    float bx[4] = {0.f, 0.f, 0.f, 0.f};
    float cls = 0.f;
    if (valid) {
      const float* bp = nmsBox + ((size_t)b * NCAND + id) * 4;
      bx[0] = bp[0]; bx[1] = bp[1]; bx[2] = bp[2]; bx[3] = bp[3];
      cls = (float)(id / MAXPC);
    }
    const int o = b * MAXDET + tid;
    out[o * 4 + 0] = bx[0];
    out[o * 4 + 1] = bx[1];
    out[o * 4 + 2] = bx[2];
    out[o * 4 + 3] = bx[3];
    out[BB * MAXDET * 4 + o] = valid ? sc : 0.f;                 // scores
    out[BB * MAXDET * 4 + BB * MAXDET + o] = cls;                // classes
  }
  __syncthreads();
  if (tid == 0) {
    int cnt = 0;
    for (int i = 0; i < MAXDET; ++i) if (ssc[i] > 0.f) ++cnt;
    out[BB * MAXDET * 4 + 2 * BB * MAXDET + b] = (float)cnt;     // n_valid
  }
}

// ---------------------------------------------------------------------------
extern "C" void kernel_launch(void* const* d_in, const int* in_sizes, int n_in,
                              void* d_out, int out_size, void* d_ws, size_t ws_size,
                              hipStream_t stream) {
  const float* pred    = (const float*)d_in[0];   // (16, 98304, 24)
  const float* anchors = (const float*)d_in[1];   // (98304, 4)

  // workspace carve-out (~832 KB total)
  float* tkScore  = (float*)d_ws;                          // NPAIR*KPRE
  int*   tkIdx    = (int*)(tkScore + NPAIR * KPRE);        // NPAIR*KPRE
  float* nmsScore = (float*)(tkIdx + NPAIR * KPRE);        // NPAIR*50
  float* nmsBox   = nmsScore + NPAIR * MAXPC;              // NPAIR*50*4

  topk_kernel<<<NPAIR, 256, 65536 * sizeof(unsigned), stream>>>(
      pred, tkScore, tkIdx);
  nms_kernel<<<NPAIR, 256, 0, stream>>>(
      pred, anchors, tkScore, tkIdx, nmsScore, nmsBox);
  combine_kernel<<<BB, 1024, 0, stream>>>(
      nmsScore, nmsBox, (float*)d_out);
}